// DihedralToCartesian_74268574482977
// MI455X (gfx1250) — compile-verified
//
#include <hip/hip_runtime.h>

// DihedralToCartesian (NeRF chain extension) for gfx1250.
// B=65536 rows, each a sequential 126-step scan -> one thread per row.
// All global traffic staged through LDS with gfx1250 async LDS ops.

#define TPB   256            // threads (= rows) per block, 8 wave32s
#define NRES  126            // steps per row
#define CH    9              // steps per chunk (multiple of 3 -> static n%3)
#define NCH   (NRES / CH)    // 14 chunks
#define ROWA  (2 * NRES)     // 252 floats per angles row
#define ROWO  (3 * NRES)     // 378 floats per output row

#if __has_builtin(__builtin_amdgcn_global_load_async_to_lds_b32)
#define HAVE_ALD 1
#else
#define HAVE_ALD 0
#endif
#if __has_builtin(__builtin_amdgcn_global_store_async_from_lds_b32)
#define HAVE_AST 1
#else
#define HAVE_AST 0
#endif

// Builtins take address-space-qualified int pointers (per hipcc diagnostics:
// param type is AS(1)/AS(3) 'int *').
typedef __attribute__((address_space(1))) int gint;
typedef __attribute__((address_space(3))) int lint;

__device__ __forceinline__ void wait_async0() {
#if HAVE_ALD || HAVE_AST
#if __has_builtin(__builtin_amdgcn_s_wait_asynccnt)
  __builtin_amdgcn_s_wait_asynccnt(0);
#else
  asm volatile("s_wait_asynccnt 0" ::: "memory");
#endif
#endif
}

__global__ __launch_bounds__(TPB) void nerf_scan_kernel(
    const float* __restrict__ angles,   // [B, 252]
    const float* __restrict__ prev,     // [B, 3, 3]
    float* __restrict__ out)            // [B, 126, 3]
{
  __shared__ float sS[2][TPB * CH];       // sin tiles (double buffered)
  __shared__ float sC[2][TPB * CH];       // cos tiles (double buffered)
  __shared__ float sO[TPB * CH * 3];      // output tile

  const int tid = threadIdx.x;
  const int b0  = blockIdx.x * TPB;

  // ---- cooperative tile loader: angles[b0..b0+255, n0..n0+CH) sin & cos ----
  auto load_tiles = [&](int k) {
    const int n0  = k * CH;
    float* dstS = sS[k & 1];
    float* dstC = sC[k & 1];
#pragma unroll
    for (int i = 0; i < CH; ++i) {
      int idx = tid + TPB * i;          // 0 .. TPB*CH-1, row-major (row,col)
      int row = idx / CH;
      int col = idx - row * CH;
      const float* gs = angles + (b0 + row) * ROWA + n0 + col;  // sin
      const float* gc = gs + NRES;                              // cos
#if HAVE_ALD
      __builtin_amdgcn_global_load_async_to_lds_b32((gint*)gs, (lint*)&dstS[idx], 0, 0);
      __builtin_amdgcn_global_load_async_to_lds_b32((gint*)gc, (lint*)&dstC[idx], 0, 0);
#else
      dstS[idx] = *gs;
      dstC[idx] = *gc;
#endif
    }
  };

  // Kick off tile 0 while we stage prev_three through sO (coalesced).
  load_tiles(0);
#pragma unroll
  for (int i = 0; i < 9; ++i) {         // 256 rows * 9 floats = 2304 <= |sO|
    int idx = tid + TPB * i;
    sO[idx] = prev[b0 * 9 + idx];
  }
  __syncthreads();                      // prev staged
  float ax = sO[tid * 9 + 0], ay = sO[tid * 9 + 1], az = sO[tid * 9 + 2];
  float bx = sO[tid * 9 + 3], by = sO[tid * 9 + 4], bz = sO[tid * 9 + 5];
  float cx = sO[tid * 9 + 6], cy = sO[tid * 9 + 7], cz = sO[tid * 9 + 8];

  // rotation constants (constant-folded by the compiler)
  float R0c[3], SBc[3];
  R0c[0] = 1.329f * cosf(2.028f);  SBc[0] = 1.329f * sinf(2.028f);
  R0c[1] = 1.458f * cosf(2.124f);  SBc[1] = 1.458f * sinf(2.124f);
  R0c[2] = 1.523f * cosf(1.941f);  SBc[2] = 1.523f * sinf(1.941f);

  wait_async0();                        // tile 0 landed in LDS
  __syncthreads();                      // also guards sO reuse below

  for (int k = 0; k < NCH; ++k) {
    if (k + 1 < NCH) load_tiles(k + 1); // prefetch next tile (other buffer)

    const float* ls = &sS[k & 1][tid * CH];
    const float* lc = &sC[k & 1][tid * CH];
    float* lo = &sO[tid * CH * 3];

#pragma unroll
    for (int c = 0; c < CH; ++c) {
      float s  = ls[c];
      float co = lc[c];
      // normalize (sin,cos): / sqrt(s^2 + c^2 + 1e-8)
      float ln = __builtin_amdgcn_rsqf(fmaf(s, s, fmaf(co, co, 1e-8f)));
      float st = s * ln, ct = co * ln;
      const int j = c % 3;              // compile-time after unroll (CH%3==0)
      float r0 = R0c[j];
      float r1 = SBc[j] * ct;
      float r2 = -SBc[j] * st;

      // bc = normalize(b - c + 1e-8)
      float ux = bx - cx + 1e-8f;
      float uy = by - cy + 1e-8f;
      float uz = bz - cz + 1e-8f;
      float us = fmaf(ux, ux, fmaf(uy, uy, uz * uz));
      float ui = __builtin_amdgcn_rsqf(fmaxf(us, 1e-24f));  // == /max(||.||,1e-12)
      float ex = ux * ui, ey = uy * ui, ez = uz * ui;

      // n = normalize(cross(b - a, bc) + 1e-8)
      float px = bx - ax, py = by - ay, pz = bz - az;
      float nx = fmaf(py, ez, -pz * ey) + 1e-8f;
      float ny = fmaf(pz, ex, -px * ez) + 1e-8f;
      float nz = fmaf(px, ey, -py * ex) + 1e-8f;
      float ns = fmaf(nx, nx, fmaf(ny, ny, nz * nz));
      float ni = __builtin_amdgcn_rsqf(fmaxf(ns, 1e-24f));
      nx *= ni; ny *= ni; nz *= ni;

      // m1 = cross(n, bc)
      float mx = fmaf(ny, ez, -nz * ey);
      float my = fmaf(nz, ex, -nx * ez);
      float mz = fmaf(nx, ey, -ny * ex);

      // d = c + r0*bc + r1*m1 + r2*n
      float dx = fmaf(r0, ex, fmaf(r1, mx, fmaf(r2, nx, cx)));
      float dy = fmaf(r0, ey, fmaf(r1, my, fmaf(r2, ny, cy)));
      float dz = fmaf(r0, ez, fmaf(r1, mz, fmaf(r2, nz, cz)));

      lo[3 * c + 0] = dx; lo[3 * c + 1] = dy; lo[3 * c + 2] = dz;

      ax = bx; ay = by; az = bz;
      bx = cx; by = cy; bz = cz;
      cx = dx; cy = dy; cz = dz;
    }

    __syncthreads();                    // output tile complete in LDS

    const int n0 = k * CH;
#pragma unroll
    for (int i = 0; i < 3 * CH; ++i) {  // 27 coalesced b32 per thread
      int idx = tid + TPB * i;
      int row = idx / (3 * CH);
      int col = idx - row * (3 * CH);
      float* g = out + (b0 + row) * ROWO + n0 * 3 + col;
#if HAVE_AST
      __builtin_amdgcn_global_store_async_from_lds_b32((gint*)g, (lint*)&sO[idx], 0, 0);
#else
      *g = sO[idx];
#endif
    }

    wait_async0();                      // next input tile + output stores done
    __syncthreads();                    // safe to reuse sO / swap buffers
  }
}

extern "C" void kernel_launch(void* const* d_in, const int* in_sizes, int n_in,
                              void* d_out, int out_size, void* d_ws, size_t ws_size,
                              hipStream_t stream) {
  const float* angles = (const float*)d_in[0];   // [B, 252] f32
  const float* prev   = (const float*)d_in[1];   // [B, 3, 3] f32
  float* out          = (float*)d_out;           // [B, 126, 3] f32
  (void)d_ws; (void)ws_size; (void)n_in; (void)out_size;

  int B = in_sizes[0] / ROWA;                    // 65536
  int grid = B / TPB;                            // 256 blocks of 8 waves
  nerf_scan_kernel<<<dim3(grid), dim3(TPB), 0, stream>>>(angles, prev, out);
}